// SmoothingTimeSeries_69114613729616
// MI455X (gfx1250) — compile-verified
//
#include <hip/hip_runtime.h>
#include <math.h>

#define T_LEN 2048
#define XDIM  64
#define YDIM  512
#define NN    4096   // XDIM*XDIM
#define LD    66     // padded LDS stride for 64x64 tiles (cols 64,65 = pad/pivot stash)

typedef __attribute__((ext_vector_type(2))) float v2f;
typedef __attribute__((ext_vector_type(8))) float v8f;

// CDNA5 f32 WMMA: D(16x16) = A(16x4) * B(4x16) + C, full f32 precision.
__device__ __forceinline__ v8f wmma4(v2f a, v2f b, v8f c) {
  return __builtin_amdgcn_wmma_f32_16x16x4_f32(false, a, false, b, (short)0, c, false, false);
}

// Operand fetch from a 64x64 LDS tile (stride LD).
// ldsN: element pair (x, kk+2h / +1) read from row x  -> contiguous b64 load.
// ldsT: element pair (kk+2h / +1, x) read down column -> two strided b32 loads.
__device__ __forceinline__ v2f ldsN(const float* P, int x, int kk, int h) {
  return *(const v2f*)(P + x * LD + kk + 2 * h);
}
__device__ __forceinline__ v2f ldsT(const float* P, int x, int kk, int h) {
  v2f r; r.x = P[(kk + 2 * h) * LD + x]; r.y = P[(kk + 2 * h + 1) * LD + x]; return r;
}

// One 16x16 output tile of a 64x64x64 product, K swept by 16 wmma steps.
//   <false,false>: A @ B^T      <false,true>: A @ B
//   <true, true>:  A^T @ B      <true, false>: A^T @ B^T
template <bool AT, bool BSTORED>
__device__ __forceinline__ v8f mm_tile64(const float* A, const float* B,
                                         int tm, int tn, int lane) {
  int nm = lane & 15, h = lane >> 4;
  int m = tm * 16 + nm, n = tn * 16 + nm;
  v8f acc = {};
#pragma unroll
  for (int kk = 0; kk < 64; kk += 4) {
    v2f a = AT ? ldsT(A, m, kk, h) : ldsN(A, m, kk, h);
    v2f b = BSTORED ? ldsT(B, n, kk, h) : ldsN(B, n, kk, h);
    acc = wmma4(a, b, acc);
  }
  return acc;
}

// ---------------------------------------------------------------------------
// K1: postX = Data @ Wmu + bmu   (2048x512 @ 512x64)
// ---------------------------------------------------------------------------
__global__ void k_postx(const float* __restrict__ Data, const float* __restrict__ Wmu,
                        const float* __restrict__ bmu, float* __restrict__ out) {
  int wave = threadIdx.x >> 5;
  int lane = threadIdx.x & 31;
  int tile = blockIdx.x * 8 + wave;
  int mt = tile >> 2, nt = tile & 3;
  int m0 = mt * 16, n0 = nt * 16;
  int nm = lane & 15, h = lane >> 4;
  v8f acc = {};
  const float* arow = Data + (size_t)(m0 + nm) * YDIM;
  for (int k = 0; k < YDIM; k += 4) {
    v2f a = *(const v2f*)(arow + k + 2 * h);
    v2f b;
    b.x = Wmu[(size_t)(k + 2 * h) * XDIM + n0 + nm];
    b.y = Wmu[(size_t)(k + 2 * h + 1) * XDIM + n0 + nm];
    acc = wmma4(a, b, acc);
  }
  float bias = bmu[n0 + nm];
#pragma unroll
  for (int v = 0; v < 8; ++v) {
    int m = m0 + v + 8 * h;
    out[(size_t)m * XDIM + n0 + nm] = acc[v] + bias;
  }
}

// ---------------------------------------------------------------------------
// K2: Cout = Data @ W + bias (+ I per 64x64 block if addEye)
// ---------------------------------------------------------------------------
__global__ void k_gemm_big(const float* __restrict__ Data, const float* __restrict__ W,
                           const float* __restrict__ bias, float* __restrict__ Cout,
                           int Trows, int addEye) {
  __shared__ __align__(16) float As[64][34];
  __shared__ __align__(16) float Bs[32][66];
  int tid = threadIdx.x;
  int wave = tid >> 5, lane = tid & 31;
  int nm = lane & 15, h = lane >> 4;
  int m0 = blockIdx.y * 64, n0 = blockIdx.x * 64;
  int tm = wave >> 1;        // m-tile 0..3
  int tn2 = (wave & 1) * 2;  // n-tile base 0 or 2
  v8f acc0 = {}, acc1 = {};
  int lr = tid >> 2, lc = (tid & 3) * 8;  // A-load: 64 rows x 32 cols
  int br = tid >> 3, bc = (tid & 7) * 8;  // B-load: 32 rows x 64 cols
  for (int kc = 0; kc < YDIM; kc += 32) {
    {
      int grow = m0 + lr; if (grow >= Trows) grow = Trows - 1;
      const float* src = Data + (size_t)grow * YDIM + kc + lc;
#pragma unroll
      for (int q = 0; q < 8; ++q) As[lr][lc + q] = src[q];
    }
    {
      const float* src = W + (size_t)(kc + br) * NN + n0 + bc;
#pragma unroll
      for (int q = 0; q < 8; ++q) Bs[br][bc + q] = src[q];
      if (kc + 32 < YDIM)
        __builtin_prefetch(W + (size_t)(kc + 32 + br) * NN + n0 + bc, 0, 1);
    }
    __syncthreads();
#pragma unroll
    for (int kk = 0; kk < 32; kk += 4) {
      v2f a = *(const v2f*)(&As[tm * 16 + nm][kk + 2 * h]);
      v2f b0, b1;
      b0.x = Bs[kk + 2 * h][tn2 * 16 + nm];
      b0.y = Bs[kk + 2 * h + 1][tn2 * 16 + nm];
      b1.x = Bs[kk + 2 * h][(tn2 + 1) * 16 + nm];
      b1.y = Bs[kk + 2 * h + 1][(tn2 + 1) * 16 + nm];
      acc0 = wmma4(a, b0, acc0);
      acc1 = wmma4(a, b1, acc1);
    }
    __syncthreads();
  }
#pragma unroll
  for (int v = 0; v < 8; ++v) {
    int m = m0 + tm * 16 + v + 8 * h;
    if (m < Trows) {
      int c0 = n0 + tn2 * 16 + nm;
      int c1 = c0 + 16;
      float r0 = acc0[v] + bias[c0];
      float r1 = acc1[v] + bias[c1];
      if (addEye) { if (c0 % 65 == 0) r0 += 1.0f; if (c1 % 65 == 0) r1 += 1.0f; }
      Cout[(size_t)m * NN + c0] = r0;
      Cout[(size_t)m * NN + c1] = r1;
    }
  }
}

// ---------------------------------------------------------------------------
// K3: per-t batched products:
//   AA[t] = AAChol[t]@AAChol[t]^T + BBChol[t-1]@BBChol[t-1]^T + 1e-6 I
//   BB[t] = AAChol[t]@BBChol[t]^T   (t < T-1)
// ---------------------------------------------------------------------------
__global__ void k_batched(const float* __restrict__ AAChol, const float* __restrict__ BBChol,
                          float* __restrict__ AA, float* __restrict__ BB) {
  __shared__ __align__(16) float Xa[64 * LD];
  __shared__ __align__(16) float Yb[64 * LD];
  __shared__ __align__(16) float Yp[64 * LD];
  int t = blockIdx.x, tid = threadIdx.x;
  for (int q = 0; q < 16; ++q) {
    int e = tid + q * 256; int i = e >> 6, j = e & 63;
    Xa[i * LD + j] = AAChol[(size_t)t * NN + e];
    Yb[i * LD + j] = (t < T_LEN - 1) ? BBChol[(size_t)t * NN + e] : 0.0f;
    Yp[i * LD + j] = (t > 0) ? BBChol[(size_t)(t - 1) * NN + e] : 0.0f;
  }
  __syncthreads();
  int wave = tid >> 5, lane = tid & 31, nm = lane & 15, h = lane >> 4;
  for (int tile = wave; tile < 16; tile += 8) {
    int tm = tile >> 2, tn = tile & 3;
    // AA tile: Xa@Xa^T + Yp@Yp^T
    v8f acc = mm_tile64<false, false>(Xa, Xa, tm, tn, lane);
#pragma unroll
    for (int kk = 0; kk < 64; kk += 4)
      acc = wmma4(ldsN(Yp, tm * 16 + nm, kk, h), ldsN(Yp, tn * 16 + nm, kk, h), acc);
#pragma unroll
    for (int v = 0; v < 8; ++v) {
      int i = tm * 16 + v + 8 * h, j = tn * 16 + nm;
      AA[(size_t)t * NN + i * 64 + j] = acc[v] + (i == j ? 1e-6f : 0.0f);
    }
    if (t < T_LEN - 1) {  // BB tile: Xa@Yb^T  (uniform branch)
      v8f accb = mm_tile64<false, false>(Xa, Yb, tm, tn, lane);
#pragma unroll
      for (int v = 0; v < 8; ++v) {
        int i = tm * 16 + v + 8 * h, j = tn * 16 + nm;
        BB[(size_t)t * NN + i * 64 + j] = accb[v];
      }
    }
  }
}

// ----------------------- in-LDS 64x64 linear algebra -----------------------
__device__ void chol64(float* M, int tid) {
  for (int i = 0; i < 64; ++i) {
    if (tid == 0) {
      float d = sqrtf(M[i * LD + i]);
      M[i * LD + i] = d;
      M[i * LD + 65] = 1.0f / d;
    }
    __syncthreads();
    float invd = M[i * LD + 65];
    if (tid > i && tid < 64) M[tid * LD + i] *= invd;
    __syncthreads();
    int r = 63 - i;
    for (int e = tid; e < r * r; e += 256) {
      int k = i + 1 + e / r, j = i + 1 + e % r;
      M[k * LD + j] -= M[k * LD + i] * M[j * LD + i];
    }
    __syncthreads();
  }
}

__device__ void trsm_lower(const float* L, float* X, int tid) {
  for (int i = 0; i < 64; ++i) {
    float invd = L[i * LD + 65];
    if (tid < 64) X[i * LD + tid] *= invd;
    __syncthreads();
    int col = tid & 63, rg = tid >> 6;
    for (int k = i + 1 + rg; k < 64; k += 4)
      X[k * LD + col] -= L[k * LD + i] * X[i * LD + col];
    __syncthreads();
  }
}

__device__ void gj_solve(float* M, float* R, float* fac, int tid) {
  for (int i = 0; i < 64; ++i) {
    if (tid == 0) M[i * LD + 65] = 1.0f / M[i * LD + i];
    __syncthreads();
    float pv = M[i * LD + 65];
    if (tid < 64) M[i * LD + tid] *= pv;
    else if (tid < 128) R[i * LD + (tid - 64)] *= pv;
    __syncthreads();
    if (tid < 64 && tid != i) fac[tid] = M[tid * LD + i];
    __syncthreads();
    for (int e = tid; e < 63 * 128; e += 256) {
      int r = e >> 7; int k = r + (r >= i);
      int c = e & 127;
      if (c < 64) M[k * LD + c] -= fac[k] * M[i * LD + c];
      else        R[k * LD + (c - 64)] -= fac[k] * R[i * LD + (c - 64)];
    }
    __syncthreads();
  }
}

// ---------------------------------------------------------------------------
// K4: the three sequential scans as 3 concurrent workgroups.
// ---------------------------------------------------------------------------
__global__ void k_scans(const float* __restrict__ AA, const float* __restrict__ BB,
                        float* __restrict__ outR, float* __restrict__ outC,
                        float* __restrict__ outS, float* __restrict__ Delta,
                        float* __restrict__ outLogdet) {
  __shared__ __align__(16) float buf0[64 * LD];
  __shared__ __align__(16) float buf1[64 * LD];
  __shared__ __align__(16) float buf2[64 * LD];
  __shared__ float fac[64];
  int tid = threadIdx.x;
  int wave = tid >> 5, lane = tid & 31, nm = lane & 15, h = lane >> 4;

  if (blockIdx.x == 0) {  // -------- Cholesky scan --------
    float* Lp = buf0; float* Ln = buf1; float* X = buf2;
    float logacc = 0.0f;
    for (int q = 0; q < 16; ++q) { int e = tid + q * 256; Ln[(e >> 6) * LD + (e & 63)] = AA[e]; }
    __syncthreads();
    chol64(Ln, tid);
    for (int q = 0; q < 16; ++q) { int e = tid + q * 256; int i = e >> 6, j = e & 63;
      outR[e] = (j <= i) ? Ln[i * LD + j] : 0.0f; }
    if (tid == 0) for (int i = 0; i < 64; ++i) logacc += logf(Ln[i * LD + i]);
    { float* t_ = Lp; Lp = Ln; Ln = t_; }
    for (int t = 1; t < T_LEN; ++t) {
      __syncthreads();
      for (int q = 0; q < 16; ++q) { int e = tid + q * 256;
        X[(e >> 6) * LD + (e & 63)] = BB[(size_t)(t - 1) * NN + e]; }
      __syncthreads();
      trsm_lower(Lp, X, tid);                       // X = Lprev^{-1} B_{t-1}
      for (int q = 0; q < 16; ++q) { int e = tid + q * 256; int i = e >> 6, j = e & 63;
        outC[(size_t)(t - 1) * NN + e] = X[j * LD + i]; }   // C = X^T
      for (int q = 0; q < 16; ++q) { int e = tid + q * 256;
        Ln[(e >> 6) * LD + (e & 63)] = AA[(size_t)t * NN + e]; }
      __syncthreads();
      for (int tile = wave; tile < 16; tile += 8) {   // Ln -= X^T X  (WMMA)
        int tm = tile >> 2, tn = tile & 3;
        v8f acc = mm_tile64<true, true>(X, X, tm, tn, lane);
#pragma unroll
        for (int v = 0; v < 8; ++v) {
          int i = tm * 16 + v + 8 * h, j = tn * 16 + nm;
          Ln[i * LD + j] -= acc[v];
        }
      }
      __syncthreads();
      chol64(Ln, tid);
      for (int q = 0; q < 16; ++q) { int e = tid + q * 256; int i = e >> 6, j = e & 63;
        outR[(size_t)t * NN + e] = (j <= i) ? Ln[i * LD + j] : 0.0f; }
      if (tid == 0) for (int i = 0; i < 64; ++i) logacc += logf(Ln[i * LD + i]);
      float* t_ = Lp; Lp = Ln; Ln = t_;
    }
    if (tid == 0) *outLogdet = -2.0f * logacc;

  } else if (blockIdx.x == 1) {  // -------- backward S scan --------
    float* M = buf0; float* Rm = buf1; float* carry = buf2;
    for (int q = 0; q < 16; ++q) { int e = tid + q * 256; carry[(e >> 6) * LD + (e & 63)] = 0.0f; }
    __syncthreads();
    for (int t = T_LEN - 2; t >= 0; --t) {
      for (int q = 0; q < 16; ++q) {
        int e = tid + q * 256; int i = e >> 6, j = e & 63;
        M[i * LD + j] = AA[(size_t)(t + 1) * NN + e] + carry[i * LD + j];
        Rm[i * LD + j] = -BB[(size_t)t * NN + j * 64 + i];   // -B_t^T
      }
      __syncthreads();
      gj_solve(M, Rm, fac, tid);                    // Rm = S_t
      for (int q = 0; q < 16; ++q) { int e = tid + q * 256;
        outS[(size_t)t * NN + e] = Rm[(e >> 6) * LD + (e & 63)]; }
      for (int q = 0; q < 16; ++q) { int e = tid + q * 256;
        M[(e >> 6) * LD + (e & 63)] = BB[(size_t)t * NN + e]; }  // stage B_t
      __syncthreads();
      for (int tile = wave; tile < 16; tile += 8) {   // carry = B_t @ S_t (WMMA)
        int tm = tile >> 2, tn = tile & 3;
        v8f acc = mm_tile64<false, true>(M, Rm, tm, tn, lane);
#pragma unroll
        for (int v = 0; v < 8; ++v) {
          int i = tm * 16 + v + 8 * h, j = tn * 16 + nm;
          carry[i * LD + j] = acc[v];
        }
      }
      __syncthreads();
    }

  } else {  // -------- forward Delta scan --------
    float* D = buf0; float* Y = buf1; float* W2 = buf2;
    for (int q = 0; q < 16; ++q) { int e = tid + q * 256; float v = AA[e];
      D[(e >> 6) * LD + (e & 63)] = v; Delta[e] = v; }
    __syncthreads();
    for (int t = 1; t < T_LEN; ++t) {
      for (int q = 0; q < 16; ++q) { int e = tid + q * 256;
        Y[(e >> 6) * LD + (e & 63)] = BB[(size_t)(t - 1) * NN + e]; }
      __syncthreads();
      gj_solve(D, Y, fac, tid);                     // Y = Dprev^{-1} B_{t-1}
      for (int q = 0; q < 16; ++q) { int e = tid + q * 256;
        D[(e >> 6) * LD + (e & 63)] = BB[(size_t)(t - 1) * NN + e]; }  // stage B
      __syncthreads();
      for (int tile = wave; tile < 16; tile += 8) {   // D_t = A_t - B^T Y (WMMA)
        int tm = tile >> 2, tn = tile & 3;
        v8f acc = mm_tile64<true, true>(D, Y, tm, tn, lane);
#pragma unroll
        for (int v = 0; v < 8; ++v) {
          int i = tm * 16 + v + 8 * h, j = tn * 16 + nm;
          float val = AA[(size_t)t * NN + i * 64 + j] - acc[v];
          W2[i * LD + j] = val;
          Delta[(size_t)t * NN + i * 64 + j] = val;
        }
      }
      __syncthreads();
      float* t_ = D; D = W2; W2 = t_;
    }
  }
}

// ---------------------------------------------------------------------------
// K5: per-t  V = inv(Delta + BB@S), VV = V @ S^T  (fully parallel over t)
// Delta region aliases V region: reads complete (into LDS) before writes.
// ---------------------------------------------------------------------------
__global__ void k_vvv(const float* __restrict__ BB, const float* __restrict__ Delta,
                      const float* __restrict__ S, float* __restrict__ V,
                      float* __restrict__ VV) {
  __shared__ __align__(16) float M[64 * LD];
  __shared__ __align__(16) float R[64 * LD];
  __shared__ __align__(16) float Sb[64 * LD];
  __shared__ float fac[64];
  int t = blockIdx.x, tid = threadIdx.x;
  int wave = tid >> 5, lane = tid & 31, nm = lane & 15, h = lane >> 4;
  bool hasS = (t < T_LEN - 1);
  for (int q = 0; q < 16; ++q) {
    int e = tid + q * 256; int i = e >> 6, j = e & 63;
    Sb[i * LD + j] = hasS ? S[(size_t)t * NN + e] : 0.0f;
    R[i * LD + j] = hasS ? BB[(size_t)t * NN + e] : 0.0f;  // stage BB
  }
  __syncthreads();
  if (hasS) {
    for (int tile = wave; tile < 16; tile += 8) {   // M = Delta + BB@S (WMMA)
      int tm = tile >> 2, tn = tile & 3;
      v8f acc = mm_tile64<false, true>(R, Sb, tm, tn, lane);
#pragma unroll
      for (int v = 0; v < 8; ++v) {
        int i = tm * 16 + v + 8 * h, j = tn * 16 + nm;
        M[i * LD + j] = Delta[(size_t)t * NN + i * 64 + j] + acc[v];
      }
    }
  } else {
    for (int q = 0; q < 16; ++q) { int e = tid + q * 256;
      M[(e >> 6) * LD + (e & 63)] = Delta[(size_t)t * NN + e]; }
  }
  __syncthreads();
  for (int q = 0; q < 16; ++q) { int e = tid + q * 256; int i = e >> 6, j = e & 63;
    R[i * LD + j] = (i == j) ? 1.0f : 0.0f; }
  __syncthreads();
  gj_solve(M, R, fac, tid);                          // R = V_t
  for (int q = 0; q < 16; ++q) { int e = tid + q * 256;
    V[(size_t)t * NN + e] = R[(e >> 6) * LD + (e & 63)]; }
  if (hasS) {
    for (int tile = wave; tile < 16; tile += 8) {    // VV = V @ S^T (WMMA)
      int tm = tile >> 2, tn = tile & 3;
      v8f acc = mm_tile64<false, false>(R, Sb, tm, tn, lane);
#pragma unroll
      for (int v = 0; v < 8; ++v) {
        int i = tm * 16 + v + 8 * h, j = tn * 16 + nm;
        VV[(size_t)t * NN + i * 64 + j] = acc[v];
      }
    }
  }
}

// ---------------------------------------------------------------------------
extern "C" void kernel_launch(void* const* d_in, const int* in_sizes, int n_in,
                              void* d_out, int out_size, void* d_ws, size_t ws_size,
                              hipStream_t stream) {
  (void)in_sizes; (void)n_in; (void)out_size; (void)ws_size;
  const float* Data = (const float*)d_in[0];
  const float* Wmu  = (const float*)d_in[1];
  const float* bmu  = (const float*)d_in[2];
  const float* WL   = (const float*)d_in[3];
  const float* bL   = (const float*)d_in[4];
  const float* WLX  = (const float*)d_in[5];
  const float* bLX  = (const float*)d_in[6];

  float* out = (float*)d_out;
  float* postX = out;
  size_t off = (size_t)T_LEN * XDIM;
  float* R  = out + off; off += (size_t)T_LEN * NN;
  float* C  = out + off; off += (size_t)(T_LEN - 1) * NN;
  float* V  = out + off; off += (size_t)T_LEN * NN;
  float* VV = out + off; off += (size_t)(T_LEN - 1) * NN;
  float* S  = out + off; off += (size_t)(T_LEN - 1) * NN;
  float* logdet = out + off;

  float* ws = (float*)d_ws;                 // 4 x 32MB workspace
  float* AAChol = ws;
  float* BBChol = AAChol + (size_t)T_LEN * NN;
  float* AAm    = BBChol + (size_t)T_LEN * NN;
  float* BBm    = AAm + (size_t)T_LEN * NN;
  float* Delta  = V;                        // stage Delta inside out-V region

  k_postx<<<64, 256, 0, stream>>>(Data, Wmu, bmu, postX);
  dim3 g2(64, 32);
  k_gemm_big<<<g2, 256, 0, stream>>>(Data, WL, bL, AAChol, T_LEN, 1);
  k_gemm_big<<<g2, 256, 0, stream>>>(Data, WLX, bLX, BBChol, T_LEN - 1, 0);
  k_batched<<<T_LEN, 256, 0, stream>>>(AAChol, BBChol, AAm, BBm);
  k_scans<<<3, 256, 0, stream>>>(AAm, BBm, R, C, S, Delta, logdet);
  k_vvv<<<T_LEN, 256, 0, stream>>>(BBm, Delta, S, V, VV);
}